// StructureAwareBottleneck_5952824672961
// MI455X (gfx1250) — compile-verified
//
#include <hip/hip_runtime.h>
#include <hip/hip_bf16.h>
#include <math.h>

typedef float v2f __attribute__((ext_vector_type(2)));
typedef float v8f __attribute__((ext_vector_type(8)));

// ---------------------------------------------------------------------------
// fp32 WMMA: D(16x16) = A(16x4) * B(4x16) + C, full fp32 (matches reference
// numerics up to reassociation).  Lane L: mr = L%16, kg = L/16.
//   A frag (v2f): a[0]=A[mr][kg*2], a[1]=A[mr][kg*2+1]
//   B frag (v2f): b[0]=B[kg*2][mr], b[1]=B[kg*2+1][mr]
//   C/D (v8f):    c[r] = C[r + 8*kg][mr]
// ---------------------------------------------------------------------------
__device__ __forceinline__ v8f wmma4(v2f a, v2f b, v8f c) {
  return __builtin_amdgcn_wmma_f32_16x16x4_f32(false, a, false, b, (short)0, c,
                                               false, false);
}

__device__ __forceinline__ float gelu_exact(float x) {
  return 0.5f * x * (1.0f + erff(x * 0.70710678118654752f));
}

// block-wide reductions, blockDim.x == 256
__device__ __forceinline__ float blk_sum(float v, float* red) {
  int t = threadIdx.x;
  red[t] = v; __syncthreads();
  for (int o = 128; o > 0; o >>= 1) {
    if (t < o) red[t] += red[t + o];
    __syncthreads();
  }
  float r = red[0]; __syncthreads();
  return r;
}
__device__ __forceinline__ float blk_max(float v, float* red) {
  int t = threadIdx.x;
  red[t] = v; __syncthreads();
  for (int o = 128; o > 0; o >>= 1) {
    if (t < o) red[t] = fmaxf(red[t], red[t + o]);
    __syncthreads();
  }
  float r = red[0]; __syncthreads();
  return r;
}
__device__ __forceinline__ float blk_min(float v, float* red) {
  int t = threadIdx.x;
  red[t] = v; __syncthreads();
  for (int o = 128; o > 0; o >>= 1) {
    if (t < o) red[t] = fminf(red[t], red[t + o]);
    __syncthreads();
  }
  float r = red[0]; __syncthreads();
  return r;
}

// ---------------------------------------------------------------------------
// Global-branch mean over (C,D,H,W) per batch
// ---------------------------------------------------------------------------
__global__ __launch_bounds__(256) void k_gmean(const float* __restrict__ x,
                                               float* __restrict__ gm) {
  __shared__ float red[256];
  int b = blockIdx.x;
  const float* xb = x + (size_t)b * 55296;
  float s = 0.f;
  for (int i = threadIdx.x; i < 55296; i += 256) s += xb[i];
  float tot = blk_sum(s, red);
  if (threadIdx.x == 0) gm[b] = tot / 55296.0f;
}

// ---------------------------------------------------------------------------
// Entire global branch (tiny): one block per batch, thread = channel.
// Rank-1 score => instance-norm stats factor; softmax is shift-invariant so
// only 1/sigma and row-max of the raw score are needed.
// ---------------------------------------------------------------------------
__global__ __launch_bounds__(256) void k_global(
    const float* __restrict__ gm,
    const float* __restrict__ tok_w, const float* __restrict__ tok_b,
    const float* __restrict__ ln1_g, const float* __restrict__ ln1_b,
    const float* __restrict__ wq, const float* __restrict__ wk,
    const float* __restrict__ wv, const float* __restrict__ wo,
    const float* __restrict__ ln2_g, const float* __restrict__ ln2_b,
    const float* __restrict__ mo_w, const float* __restrict__ mo_b,
    float* __restrict__ gvec) {
  __shared__ float red[256];
  __shared__ float t_s[256];
  __shared__ float q_s[1024], k_s[1024], v_s[1024], o_s[1024];
  int b = blockIdx.x, c = threadIdx.x;
  float m = gm[b];
  float tok = m * tok_w[c] + tok_b[c];
  float resv = tok;
  float S1 = blk_sum(tok, red);
  float S2 = blk_sum(tok * tok, red);
  float mn = S1 * (1.f / 256.f);
  float var = S2 * (1.f / 256.f) - mn * mn;
  t_s[c] = (tok - mn) * rsqrtf(var + 1e-6f) * ln1_g[c] + ln1_b[c];
  __syncthreads();
  // q/k/v projections: 1x256 @ 256x1024 each
  for (int rep = 0; rep < 4; ++rep) {
    int o = rep * 256 + c;
    float aq = 0.f, ak = 0.f, av = 0.f;
    for (int i = 0; i < 256; ++i) {
      float ti = t_s[i];
      aq += ti * wq[i * 1024 + o];
      ak += ti * wk[i * 1024 + o];
      av += ti * wv[i * 1024 + o];
    }
    q_s[o] = aq; k_s[o] = ak; v_s[o] = av;
  }
  __syncthreads();
  for (int h = 0; h < 4; ++h) {           // heads OUTER (reshape B,Hh,C)
    float qi = q_s[h * 256 + c], ki = k_s[h * 256 + c];
    float Sq  = blk_sum(qi, red),       Sq2 = blk_sum(qi * qi, red);
    float Sk  = blk_sum(ki, red),       Sk2 = blk_sum(ki * ki, red);
    float kmx = blk_max(ki, red),       kmn = blk_min(ki, red);
    float mu  = (Sq * Sk)  * (1.f / 65536.f);
    float ex2 = (Sq2 * Sk2) * (1.f / 65536.f);
    float rstd = rsqrtf(ex2 - mu * mu + 1e-5f);
    float rmax = qi * (qi >= 0.f ? kmx : kmn);  // row max of raw score
    float den = 0.f, acc = 0.f;
    for (int j = 0; j < 256; ++j) {
      float pe = __expf((qi * k_s[h * 256 + j] - rmax) * rstd);
      den += pe;
      acc += pe * v_s[h * 256 + j];
    }
    o_s[h * 256 + c] = acc / den;
    __syncthreads();
  }
  // out = o^T(flattened c-outer) @ wo (1024x256)
  float outv = 0.f;
  for (int i = 0; i < 256; ++i) {
    const float* w = wo + (size_t)(i * 4) * 256 + c;
    outv += o_s[i] * w[0] + o_s[256 + i] * w[256] + o_s[512 + i] * w[512] +
            o_s[768 + i] * w[768];
  }
  float ov = outv + resv;
  float T1 = blk_sum(ov, red), T2 = blk_sum(ov * ov, red);
  float mn2 = T1 * (1.f / 256.f), var2 = T2 * (1.f / 256.f) - mn2 * mn2;
  float tok2 = (ov - mn2) * rsqrtf(var2 + 1e-6f) * ln2_g[c] + ln2_b[c];
  t_s[c] = tok2;
  __syncthreads();
  float a2 = mo_b[c];
  for (int i = 0; i < 256; ++i) a2 += t_s[i] * mo_w[c * 256 + i];  // 'bc,oc->bo'
  gvec[b * 256 + c] = gelu_exact(a2);
}

// ---------------------------------------------------------------------------
// Local branch: patchify (2x2x2) + 8x8 embed + LN over C.
// Emits res (B,216,C) and eln rows (bp*8+pv, C).
// ---------------------------------------------------------------------------
__global__ __launch_bounds__(256) void k_local_embed(
    const float* __restrict__ x, const float* __restrict__ pw,
    const float* __restrict__ pb, const float* __restrict__ ln_g,
    const float* __restrict__ ln_b, float* __restrict__ res,
    float* __restrict__ eln) {
  __shared__ float pw_s[64];
  __shared__ float pb_s[8];
  __shared__ float sc[8 * 256];
  __shared__ float part1[256], part2[256];
  __shared__ float mean_s[8], rstd_s[8];
  int bp = blockIdx.x, b = bp / 27, p = bp % 27;
  int d = p / 9, hh = (p / 3) % 3, w = p % 3;
  int c = threadIdx.x;
  if (c < 64) pw_s[c] = pw[c];
  if (c < 8) pb_s[c] = pb[c];
  __syncthreads();
  const float* xb = x + (size_t)(b * 256 + c) * 216;
  float patch[8];
#pragma unroll
  for (int u = 0; u < 8; ++u) {
    int p1 = u >> 2, p2 = (u >> 1) & 1, p3 = u & 1;
    patch[u] = xb[(d * 2 + p1) * 36 + (hh * 2 + p2) * 6 + (w * 2 + p3)];
  }
  float tokv[8];
#pragma unroll
  for (int e = 0; e < 8; ++e) {
    float a = pb_s[e];
#pragma unroll
    for (int u = 0; u < 8; ++u) a += pw_s[e * 8 + u] * patch[u];
    tokv[e] = a;
    sc[e * 256 + c] = a;
    int p1 = e >> 2, p2 = (e >> 1) & 1, p3 = e & 1;
    int s = (d * 2 + p1) * 36 + (hh * 2 + p2) * 6 + (w * 2 + p3);
    res[(size_t)(b * 216 + s) * 256 + c] = a;   // residual uses raw tok
  }
  __syncthreads();
  int g = c >> 5, lane = c & 31;
  float s1 = 0.f, s2 = 0.f;
#pragma unroll
  for (int k = 0; k < 8; ++k) {
    float vv = sc[g * 256 + lane + k * 32];
    s1 += vv; s2 += vv * vv;
  }
  part1[c] = s1; part2[c] = s2;
  __syncthreads();
  if (c < 8) {
    float a = 0.f, bb = 0.f;
    for (int l = 0; l < 32; ++l) { a += part1[c * 32 + l]; bb += part2[c * 32 + l]; }
    float mn = a * (1.f / 256.f), vr = bb * (1.f / 256.f) - mn * mn;
    mean_s[c] = mn; rstd_s[c] = rsqrtf(vr + 1e-6f);
  }
  __syncthreads();
  float lg = ln_g[c], lb = ln_b[c];
#pragma unroll
  for (int e = 0; e < 8; ++e)
    eln[(size_t)(bp * 8 + e) * 256 + c] =
        (tokv[e] - mean_s[e]) * rstd_s[e] * lg + lb;
}

// ---------------------------------------------------------------------------
// fp32 WMMA GEMM: C = act(A(MxK) @ B(KxN) [+bias]).  TRANSB/ACT are template
// params so the inner loop is branch-free.  blockDim=128 (4 waves); each wave
// owns a 16(M) x 64(N) strip; all 4 waves share one n-strip, so the 16x64 B
// tile is staged through LDS once per K-chunk (coalesced float4 loads, 4x
// reuse).  Requires M%64==0, N%64==0, K%16==0.
// ---------------------------------------------------------------------------
template <bool TRANSB, bool ACT>
__global__ __launch_bounds__(128) void k_gemm_wmma(
    const float* __restrict__ A, const float* __restrict__ B,
    const float* __restrict__ bias, float* __restrict__ C, int M, int N,
    int K) {
  __shared__ float Bs[16 * 64];
  int tid = threadIdx.x;
  int wave = tid >> 5, lane = tid & 31;
  int mr = lane & 15, kg = lane >> 4;
  int m0 = (blockIdx.y * 4 + wave) * 16;
  int n0 = blockIdx.x * 64;
  v8f z = {0, 0, 0, 0, 0, 0, 0, 0};
  v8f acc[4] = {z, z, z, z};
  const float* arow = A + (size_t)(m0 + mr) * K;
  for (int k0 = 0; k0 < K; k0 += 16) {
    __syncthreads();   // guard Bs against previous chunk's readers
    if (!TRANSB) {
      // rows k0..k0+15 of B, cols n0..n0+63: thread loads two float4s
      int r = tid >> 4, c4 = (tid & 15) * 4;
      float4 v0 = *(const float4*)(B + (size_t)(k0 + r) * N + n0 + c4);
      float4 v1 = *(const float4*)(B + (size_t)(k0 + r + 8) * N + n0 + c4);
      *(float4*)(Bs + r * 64 + c4) = v0;
      *(float4*)(Bs + (r + 8) * 64 + c4) = v1;
    } else {
      // B stored (N x K): float4 runs along k, scatter into Bs rows
      int c = tid & 63, q0 = tid >> 6;  // q0 in {0,1}
#pragma unroll
      for (int qq = 0; qq < 2; ++qq) {
        int q = q0 + qq * 2;
        float4 v4 = *(const float4*)(B + (size_t)(n0 + c) * K + k0 + q * 4);
        Bs[(q * 4 + 0) * 64 + c] = v4.x;
        Bs[(q * 4 + 1) * 64 + c] = v4.y;
        Bs[(q * 4 + 2) * 64 + c] = v4.z;
        Bs[(q * 4 + 3) * 64 + c] = v4.w;
      }
    }
    __syncthreads();
#pragma unroll
    for (int kc = 0; kc < 4; ++kc) {
      int ka = kc * 4 + kg * 2;
      v2f a = *(const v2f*)(arow + k0 + ka);
#pragma unroll
      for (int t = 0; t < 4; ++t) {
        int n = t * 16 + mr;
        v2f bv;
        bv[0] = Bs[ka * 64 + n];
        bv[1] = Bs[(ka + 1) * 64 + n];
        acc[t] = wmma4(a, bv, acc[t]);
      }
    }
  }
#pragma unroll
  for (int t = 0; t < 4; ++t) {
    int n = n0 + t * 16 + mr;
    float bb = ACT ? bias[n] : 0.f;
#pragma unroll
    for (int r = 0; r < 8; ++r) {
      int row = m0 + r + 8 * kg;
      float v = acc[t][r] + bb;
      if (ACT) v = gelu_exact(v);
      C[(size_t)row * N + n] = v;
    }
  }
}

// ---------------------------------------------------------------------------
// Local attention core: one block per (bp, head); 864 blocks, 8 waves.
// score(256x256) = q(256x8) @ k(8x256) via WMMA (K=8 -> 2 steps of 4).
// Pass 1: stats (sum, sum^2, row-max). Pass 2: recompute + exp + p@v fused.
// ---------------------------------------------------------------------------
__global__ __launch_bounds__(256) void k_attn_local(
    const float* __restrict__ Qb, const float* __restrict__ Kb,
    const float* __restrict__ Vb, float* __restrict__ Ofl) {
  __shared__ float q_s[256 * 8];       // q[i][m]
  __shared__ float kt_s[8 * 256];      // k[m][j]
  __shared__ float v_s[256 * 8];       // v[j][m]
  __shared__ float tbuf[8][16 * 17];   // per-wave 16x16 tile, padded pitch
  __shared__ float rowmax_s[256];
  __shared__ float red1[256], red2[256];
  __shared__ float rstd_sh;
  int bp = blockIdx.x >> 2, h = blockIdx.x & 3;
  int b = bp / 27, p = bp % 27;
  int d = p / 9, ph = (p / 3) % 3, pwv = p % 3;
  int tid = threadIdx.x;
  {
    int c = tid;
#pragma unroll
    for (int pv = 0; pv < 8; ++pv) {
      size_t idx = (size_t)(bp * 8 + pv) * 1024 + c * 4 + h;  // col = c*Hh + h
      q_s[c * 8 + pv] = Qb[idx];
      kt_s[pv * 256 + c] = Kb[idx];
      v_s[c * 8 + pv] = Vb[idx];
    }
  }
  __syncthreads();
  int wave = tid >> 5, lane = tid & 31;
  int mr = lane & 15, kg = lane >> 4;
  float* tb = tbuf[wave];
  float s1 = 0.f, s2 = 0.f;
  // ---- pass 1: raw score stats -------------------------------------------
  for (int rbi = 0; rbi < 2; ++rbi) {
    int i0 = (wave + rbi * 8) * 16;
    float rmax = -3.0e38f;
    for (int jt = 0; jt < 16; ++jt) {
      int j0 = jt * 16;
      v8f acc = {0, 0, 0, 0, 0, 0, 0, 0};
#pragma unroll
      for (int kc = 0; kc < 2; ++kc) {
        int ka = kc * 4 + kg * 2;
        v2f a, bv;
        a[0] = q_s[(i0 + mr) * 8 + ka];
        a[1] = q_s[(i0 + mr) * 8 + ka + 1];
        bv[0] = kt_s[ka * 256 + j0 + mr];
        bv[1] = kt_s[(ka + 1) * 256 + j0 + mr];
        acc = wmma4(a, bv, acc);
      }
#pragma unroll
      for (int r = 0; r < 8; ++r) {
        float vv = acc[r];
        s1 += vv; s2 += vv * vv;
        tb[(r + 8 * kg) * 17 + mr] = vv;
      }
      if (lane < 16) {  // per-wave LDS, same-wave DS ordering
        for (int j = 0; j < 16; ++j) rmax = fmaxf(rmax, tb[lane * 17 + j]);
      }
    }
    if (lane < 16) rowmax_s[i0 + lane] = rmax;
  }
  red1[tid] = s1; red2[tid] = s2;
  __syncthreads();
  for (int o = 128; o > 0; o >>= 1) {
    if (tid < o) { red1[tid] += red1[tid + o]; red2[tid] += red2[tid + o]; }
    __syncthreads();
  }
  if (tid == 0) {
    float mu = red1[0] * (1.f / 65536.f);
    float var = red2[0] * (1.f / 65536.f) - mu * mu;
    rstd_sh = rsqrtf(var + 1e-5f);   // InstanceNorm eps; mu cancels in softmax
  }
  __syncthreads();
  float rstd = rstd_sh;
  // ---- pass 2: softmax + p@v ---------------------------------------------
  for (int rbi = 0; rbi < 2; ++rbi) {
    int i0 = (wave + rbi * 8) * 16;
    float den = 0.f;
    float num[8] = {0, 0, 0, 0, 0, 0, 0, 0};
    float rmax = (lane < 16) ? rowmax_s[i0 + lane] : 0.f;
    for (int jt = 0; jt < 16; ++jt) {
      int j0 = jt * 16;
      v8f acc = {0, 0, 0, 0, 0, 0, 0, 0};
#pragma unroll
      for (int kc = 0; kc < 2; ++kc) {
        int ka = kc * 4 + kg * 2;
        v2f a, bv;
        a[0] = q_s[(i0 + mr) * 8 + ka];
        a[1] = q_s[(i0 + mr) * 8 + ka + 1];
        bv[0] = kt_s[ka * 256 + j0 + mr];
        bv[1] = kt_s[(ka + 1) * 256 + j0 + mr];
        acc = wmma4(a, bv, acc);
      }
#pragma unroll
      for (int r = 0; r < 8; ++r) tb[(r + 8 * kg) * 17 + mr] = acc[r];
      if (lane < 16) {
        for (int j = 0; j < 16; ++j) {
          float sv = tb[lane * 17 + j];
          float pe = __expf((sv - rmax) * rstd);
          den += pe;
          const float* vp = &v_s[(j0 + j) * 8];
#pragma unroll
          for (int m2 = 0; m2 < 8; ++m2) num[m2] += pe * vp[m2];
        }
      }
    }
    if (lane < 16) {
      int cc = i0 + lane;
      float inv = 1.f / den;
#pragma unroll
      for (int m2 = 0; m2 < 8; ++m2) {
        int p1 = m2 >> 2, p2 = (m2 >> 1) & 1, p3 = m2 & 1;
        int s = (d * 2 + p1) * 36 + (ph * 2 + p2) * 6 + (pwv * 2 + p3);
        Ofl[(size_t)(b * 216 + s) * 1024 + cc * 4 + h] = num[m2] * inv;
      }
    }
  }
}

// ---------------------------------------------------------------------------
// t2 = LN(out + res) per row over C
// ---------------------------------------------------------------------------
__global__ __launch_bounds__(256) void k_ln2_local(
    const float* __restrict__ outb, const float* __restrict__ res,
    const float* __restrict__ g, const float* __restrict__ bb,
    float* __restrict__ t2) {
  __shared__ float red[256];
  int row = blockIdx.x, c = threadIdx.x;
  float v = outb[(size_t)row * 256 + c] + res[(size_t)row * 256 + c];
  float S1 = blk_sum(v, red), S2 = blk_sum(v * v, red);
  float mn = S1 * (1.f / 256.f), var = S2 * (1.f / 256.f) - mn * mn;
  t2[(size_t)row * 256 + c] = (v - mn) * rsqrtf(var + 1e-6f) * g[c] + bb[c];
}

// ---------------------------------------------------------------------------
// y = x + broadcast(global) + local
// ---------------------------------------------------------------------------
__global__ __launch_bounds__(256) void k_final(const float* __restrict__ x,
                                               const float* __restrict__ gvec,
                                               const float* __restrict__ lres,
                                               float* __restrict__ y) {
  int idx = blockIdx.x * 256 + threadIdx.x;
  int c = idx & 255, bs = idx >> 8;
  int b = bs / 216, s = bs - b * 216;
  size_t xi = (size_t)(b * 256 + c) * 216 + s;
  y[xi] = x[xi] + gvec[b * 256 + c] + lres[(size_t)bs * 256 + c];
}

extern "C" void kernel_launch(void* const* d_in, const int* in_sizes, int n_in,
                              void* d_out, int out_size, void* d_ws,
                              size_t ws_size, hipStream_t stream) {
  const float* x       = (const float*)d_in[0];
  const float* g_tok_w = (const float*)d_in[1];
  const float* g_tok_b = (const float*)d_in[2];
  const float* g_ln1_g = (const float*)d_in[3];
  const float* g_ln1_b = (const float*)d_in[4];
  const float* g_wq    = (const float*)d_in[5];
  const float* g_wk    = (const float*)d_in[6];
  const float* g_wv    = (const float*)d_in[7];
  const float* g_wo    = (const float*)d_in[8];
  const float* g_ln2_g = (const float*)d_in[9];
  const float* g_ln2_b = (const float*)d_in[10];
  const float* g_mo_w  = (const float*)d_in[11];
  const float* g_mo_b  = (const float*)d_in[12];
  const float* l_pw    = (const float*)d_in[13];
  const float* l_pb    = (const float*)d_in[14];
  const float* l_ln1_g = (const float*)d_in[15];
  const float* l_ln1_b = (const float*)d_in[16];
  const float* l_wq    = (const float*)d_in[17];
  const float* l_wk    = (const float*)d_in[18];
  const float* l_wv    = (const float*)d_in[19];
  const float* l_wo    = (const float*)d_in[20];
  const float* l_ln2_g = (const float*)d_in[21];
  const float* l_ln2_b = (const float*)d_in[22];
  const float* l_mo_w  = (const float*)d_in[23];
  const float* l_mo_b  = (const float*)d_in[24];

  float* ws   = (float*)d_ws;
  float* gm   = ws;                 // 8
  float* gvec = ws + 8;             // 2048
  float* eln  = ws + 8 + 2048;      // 1728*256
  float* res  = eln + 442368;       // 1728*256
  float* Qb   = res + 442368;       // 1728*1024
  float* Kb   = Qb + 1769472;       // 1728*1024
  float* Vb   = Kb + 1769472;       // 1728*1024
  float* ofl  = Vb + 1769472;       // 1728*1024
  float* outb = Qb;                 // alias: Qb dead after attention
  float* t2   = Kb;                 // alias: Kb dead after attention
  float* lres = Vb;                 // alias: Vb dead after attention

  k_gmean<<<8, 256, 0, stream>>>(x, gm);
  k_local_embed<<<216, 256, 0, stream>>>(x, l_pw, l_pb, l_ln1_g, l_ln1_b, res,
                                         eln);
  k_gemm_wmma<false, false><<<dim3(16, 27), 128, 0, stream>>>(
      eln, l_wq, nullptr, Qb, 1728, 1024, 256);
  k_gemm_wmma<false, false><<<dim3(16, 27), 128, 0, stream>>>(
      eln, l_wk, nullptr, Kb, 1728, 1024, 256);
  k_gemm_wmma<false, false><<<dim3(16, 27), 128, 0, stream>>>(
      eln, l_wv, nullptr, Vb, 1728, 1024, 256);
  k_global<<<8, 256, 0, stream>>>(gm, g_tok_w, g_tok_b, g_ln1_g, g_ln1_b, g_wq,
                                  g_wk, g_wv, g_wo, g_ln2_g, g_ln2_b, g_mo_w,
                                  g_mo_b, gvec);
  k_attn_local<<<864, 256, 0, stream>>>(Qb, Kb, Vb, ofl);
  k_gemm_wmma<false, false><<<dim3(4, 27), 128, 0, stream>>>(
      ofl, l_wo, nullptr, outb, 1728, 256, 1024);
  k_ln2_local<<<1728, 256, 0, stream>>>(outb, res, l_ln2_g, l_ln2_b, t2);
  k_gemm_wmma<true, true><<<dim3(4, 27), 128, 0, stream>>>(
      t2, l_mo_w, l_mo_b, lres, 1728, 256, 256);
  k_final<<<1728, 256, 0, stream>>>(x, gvec, lres, (float*)d_out);
}